// ConstituencyTreeCRF_23424751632870
// MI455X (gfx1250) — compile-verified
//
#include <hip/hip_runtime.h>
#include <stdint.h>

// CKY inside algorithm (ConstituencyTreeCRF), B=1024, N=128.
//   alpha[k][s] = scores[s][s+k] + LSE_{j<k}( alpha[j][s] + alpha[k-1-j][s+j+1] )
//   out[b] = alpha[N-1][0]
//
// MI455X strategy:
//  * One workgroup per batch, 256 threads (8 wave32 waves).
//  * Whole DP table packed triangular in LDS: A[triOff(k)+s], 8256 f32 = 33 KB.
//  * scores diagonals staged INTO their alpha rows by CDNA5 async global->LDS
//    copies (per-lane LDS scatter = free row-major -> diagonal transpose,
//    coalesced global reads), then overwritten in place.
//  * Width-adaptive parallelism: P = pow2 lanes per span (P ~ BLOCK/spans,
//    capped at 32 = one wave), split loop strided by P, partial max / sum-exp
//    merged by constant-mask wave32 shfl_xor butterflies.
//  * Two-pass streaming logsumexp: one v_exp_f32 per split evaluation.
//  * All hot LDS addresses lane-consecutive -> bank-conflict free.

#define N        128
#define NTRI     8256          // N*(N+1)/2
#define BLOCK    256
#define HUGE_NEG (-1.0e9f)

__device__ __forceinline__ unsigned triOff(int k) {
  // start of packed diagonal row k (element index); quadratic, 2nd-diff = -1
  return (unsigned)((k * (2 * N - k + 1)) >> 1);
}

__global__ __launch_bounds__(BLOCK) void cky_inside_kernel(
    const float* __restrict__ scores,  // (B, N, N)
    float* __restrict__ out)           // (B,)
{
  __shared__ float A[NTRI];            // staged scores, overwritten by alpha

  const int b   = blockIdx.x;
  const int tid = threadIdx.x;
  const float* __restrict__ sb = scores + (size_t)b * N * N;

  // ---------------------------------------------------------------------------
  // Stage upper triangle: A[triOff(c-r) + r] = scores[r][c] for c >= r.
  // Half-block h handles rows r = h, h+2, ...; lanes c contiguous -> coalesced
  // global reads; per-lane LDS dest VGPR -> async scatter does the transpose.
  // ---------------------------------------------------------------------------
  {
    const int c = tid & (N - 1);
    const int rstart = tid >> 7;       // 0 or 1
    for (int r = rstart; r < N; r += 2) {
      if (c >= r) {
        unsigned lds_byte = (unsigned)(uintptr_t)(&A[triOff(c - r) + r]);
        unsigned long long ga = (unsigned long long)(uintptr_t)(sb + r * N + c);
        asm volatile("global_load_async_to_lds_b32 %0, %1, off"
                     :: "v"(lds_byte), "v"(ga)
                     : "memory");
      }
    }
  }
  asm volatile("s_wait_asynccnt 0" ::: "memory");
  __syncthreads();   // all waves' async LDS writes visible workgroup-wide

  // ---------------------------------------------------------------------------
  // DP over widths k = 1..N-1. Row k is written only by its own lane group and
  // read only by later iterations -> single barrier per width.
  // ---------------------------------------------------------------------------
  for (int k = 1; k < N; ++k) {
    const int spans = N - k;

    // logP = min(5, floor(log2(BLOCK/spans))) , branchless.
    // For BLOCK=256: logP = __clz(spans-1) - 24 (clz(0)=32 handles spans==1).
    int logP = __clz(spans - 1) - 24;
    if (logP > 5) logP = 5;
    const int P   = 1 << logP;         // >= 2 for every k (spans <= 127)
    const int P2  = P * P;
    const int sub = tid & (P - 1);
    const int s   = tid >> logP;

    if (s < spans) {
      // split j walks j0, j0+P, ... < k ; incremental triangular addressing:
      //   left  = A[triOff(j)     + s    ]  delta: +dL, dL -= P^2
      //   right = A[triOff(k-1-j) + s+j+1]  delta: -dR, dR += P^2
      const int j0 = sub;
      const unsigned aL0 = triOff(j0) + (unsigned)s;
      const unsigned aR0 = triOff(k - 1 - j0) + (unsigned)(s + j0 + 1);
      const int dL0 = P * (N - j0) - ((P * (P - 1)) >> 1);
      const int dR0 = P * (N - k + 1 + j0) + ((P * (P + 1)) >> 1) - P;

      // ---- pass 1: running max over this lane's split subset ----
      float m = HUGE_NEG;
      {
        unsigned aL = aL0, aR = aR0;
        int dL = dL0, dR = dR0;
        for (int j = j0; j < k; j += P) {
          float v = A[aL] + A[aR];
          m = fmaxf(m, v);
          aL += (unsigned)dL; dL -= P2;
          aR -= (unsigned)dR; dR += P2;
        }
      }
      // group max: P consecutive lanes of one wave, constant-mask butterflies
      m = fmaxf(m, __shfl_xor(m, 1, 32));
      if (P >= 4)  m = fmaxf(m, __shfl_xor(m, 2, 32));
      if (P >= 8)  m = fmaxf(m, __shfl_xor(m, 4, 32));
      if (P >= 16) m = fmaxf(m, __shfl_xor(m, 8, 32));
      if (P >= 32) m = fmaxf(m, __shfl_xor(m, 16, 32));

      // ---- pass 2: sum of exp(v - m); one v_exp_f32 per split ----
      float acc = 0.0f;
      {
        unsigned aL = aL0, aR = aR0;
        int dL = dL0, dR = dR0;
        for (int j = j0; j < k; j += P) {
          float v = A[aL] + A[aR];
          acc += __expf(v - m);
          aL += (unsigned)dL; dL -= P2;
          aR -= (unsigned)dR; dR += P2;
        }
      }
      acc += __shfl_xor(acc, 1, 32);
      if (P >= 4)  acc += __shfl_xor(acc, 2, 32);
      if (P >= 8)  acc += __shfl_xor(acc, 4, 32);
      if (P >= 16) acc += __shfl_xor(acc, 8, 32);
      if (P >= 32) acc += __shfl_xor(acc, 16, 32);

      if (sub == 0) {
        // staged score for span (s, s+k) sits exactly where alpha row k goes
        const unsigned w = triOff(k) + (unsigned)s;
        A[w] = m + __logf(acc) + A[w];
      }
    }
    __syncthreads();
  }

  if (tid == 0) out[b] = A[NTRI - 1];   // alpha[N-1][0]
}

extern "C" void kernel_launch(void* const* d_in, const int* in_sizes, int n_in,
                              void* d_out, int out_size, void* d_ws, size_t ws_size,
                              hipStream_t stream) {
  (void)n_in; (void)d_ws; (void)ws_size; (void)out_size;
  const float* scores = (const float*)d_in[0];
  float* out = (float*)d_out;
  const int B = in_sizes[0] / (N * N);   // 1024
  cky_inside_kernel<<<dim3(B), dim3(BLOCK), 0, stream>>>(scores, out);
}